// bspline_array_53163105189889
// MI455X (gfx1250) — compile-verified
//
#include <hip/hip_runtime.h>

#ifndef __has_builtin
#define __has_builtin(x) 0
#endif

namespace {
constexpr int kN      = 256;  // curves (block size, 8 wave32s)
constexpr int kT      = 16;   // knots / coeffs per curve
constexpr int kStride = 18;   // padded LDS row stride in dwords:
                              // 72B rows -> 8B aligned, and 18*dn mod 64 distinct
                              // for all 32 lanes of a wave -> bank-conflict-free
}

#if __has_builtin(__builtin_amdgcn_global_load_async_to_lds_b32)
#define BS_HAS_ASYNC_LDS 1
#else
#define BS_HAS_ASYNC_LDS 0
#endif

#if __has_builtin(__builtin_amdgcn_rcpf)
__device__ __forceinline__ float bs_rcp(float v) { return __builtin_amdgcn_rcpf(v); }
#else
__device__ __forceinline__ float bs_rcp(float v) { return 1.0f / v; }
#endif

typedef __attribute__((address_space(1))) int* global_i32_ptr;
typedef __attribute__((address_space(3))) int* lds_i32_ptr;

__global__ __launch_bounds__(kN) void bspline_eval_gfx1250_kernel(
    const float* __restrict__ x,
    const float* __restrict__ t,
    const float* __restrict__ c,
    float* __restrict__ out,
    int B)
{
  __shared__ float tS[kN * kStride];
  __shared__ float cS[kN * kStride];

  const int n = threadIdx.x;

  // ---- Stage coefficient table into LDS.
  // gfx1250 path: async global->LDS DMA (ASYNCcnt) overlapping the register
  // bitonic sort of the knots below.
#if BS_HAS_ASYNC_LDS
  {
    global_i32_ptr g = (global_i32_ptr)(void*)(const_cast<float*>(c) + (size_t)n * kT);
    lds_i32_ptr    l = (lds_i32_ptr)(void*)(&cS[n * kStride]);
#pragma unroll
    for (int i = 0; i < kT; ++i) {
      __builtin_amdgcn_global_load_async_to_lds_b32(g + i, l + i, 0, 0);
    }
  }
#endif

  // ---- Load & sort this thread's 16 knots entirely in registers
  // (fully unrolled bitonic network; all indices constant after unroll).
  float tt[kT];
  {
    const float4* tv = (const float4*)(t + (size_t)n * kT);
    float4 a0 = tv[0], a1 = tv[1], a2 = tv[2], a3 = tv[3];
    tt[0] = a0.x; tt[1] = a0.y; tt[2]  = a0.z; tt[3]  = a0.w;
    tt[4] = a1.x; tt[5] = a1.y; tt[6]  = a1.z; tt[7]  = a1.w;
    tt[8] = a2.x; tt[9] = a2.y; tt[10] = a2.z; tt[11] = a2.w;
    tt[12] = a3.x; tt[13] = a3.y; tt[14] = a3.z; tt[15] = a3.w;
  }
#pragma unroll
  for (int k = 2; k <= kT; k <<= 1) {
#pragma unroll
    for (int j = k >> 1; j > 0; j >>= 1) {
#pragma unroll
      for (int i = 0; i < kT; ++i) {
        const int l = i ^ j;
        if (l > i) {
          const bool up = (i & k) == 0;
          const float a = tt[i], b = tt[l];
          const float mn = fminf(a, b), mx = fmaxf(a, b);
          tt[i] = up ? mn : mx;
          tt[l] = up ? mx : mn;
        }
      }
    }
  }

  // Publish sorted knots to LDS (needed for the dynamically-indexed window).
#pragma unroll
  for (int i = 0; i < kT; ++i) tS[n * kStride + i] = tt[i];

#if BS_HAS_ASYNC_LDS
#if __has_builtin(__builtin_amdgcn_s_wait_asynccnt)
  __builtin_amdgcn_s_wait_asynccnt(0);
#else
  asm volatile("s_wait_asynccnt 0" ::: "memory");
#endif
#else
  {
    const float4* cv = (const float4*)(c + (size_t)n * kT);
    float4 a0 = cv[0], a1 = cv[1], a2 = cv[2], a3 = cv[3];
    float* dst = &cS[n * kStride];
    dst[0] = a0.x; dst[1] = a0.y; dst[2]  = a0.z; dst[3]  = a0.w;
    dst[4] = a1.x; dst[5] = a1.y; dst[6]  = a1.z; dst[7]  = a1.w;
    dst[8] = a2.x; dst[9] = a2.y; dst[10] = a2.z; dst[11] = a2.w;
    dst[12] = a3.x; dst[13] = a3.y; dst[14] = a3.z; dst[15] = a3.w;
  }
#endif
  __syncthreads();

  const float* trow = &tS[n * kStride];
  const float* crow = &cS[n * kStride];

  // ---- Persistent loop over query rows; x/out accesses fully coalesced.
  for (int b = blockIdx.x; b < B; b += (int)gridDim.x) {
    const size_t idx = (size_t)b * kN + n;
    const float xv = x[idx];

    const int bn = b + (int)gridDim.x;
    if (bn < B) __builtin_prefetch(&x[(size_t)bn * kN + n], 0, 1);

    // searchsorted(side='right') on register copy: count of t[i] <= x.
    int cnt = 0;
#pragma unroll
    for (int i = 0; i < kT; ++i) cnt += (tt[i] <= xv) ? 1 : 0;
    int m = cnt - 1;
    m = m < 3 ? 3 : (m > 11 ? 11 : m);

    // Knot window t[m-2..m+3] and coeff window c[m-3..m] from padded LDS.
    const float tm2 = trow[m - 2];
    const float tm1 = trow[m - 1];
    const float tm0 = trow[m];
    const float tp1 = trow[m + 1];
    const float tp2 = trow[m + 2];
    const float tp3 = trow[m + 3];

    float d0 = crow[m - 3];
    float d1 = crow[m - 2];
    float d2 = crow[m - 1];
    float d3 = crow[m];

    // All six alphas depend only on the knot window -> compute up front so
    // the six v_rcp_f32 issue back-to-back (ILP in the transcendental pipe);
    // only the 3-deep lerp chain below stays serial.
    const float x0 = xv - tm0;
    const float x1 = xv - tm1;
    const float x2 = xv - tm2;
    const float a13 = x0 * bs_rcp(tp3 - tm0);  // r=1, j=3
    const float a12 = x1 * bs_rcp(tp2 - tm1);  // r=1, j=2
    const float a11 = x2 * bs_rcp(tp1 - tm2);  // r=1, j=1
    const float a23 = x0 * bs_rcp(tp2 - tm0);  // r=2, j=3
    const float a22 = x1 * bs_rcp(tp1 - tm1);  // r=2, j=2
    const float a33 = x0 * bs_rcp(tp1 - tm0);  // r=3, j=3

    // de Boor, K=3, in the reference's descending-j order.
    d3 = (1.0f - a13) * d2 + a13 * d3;
    d2 = (1.0f - a12) * d1 + a12 * d2;
    d1 = (1.0f - a11) * d0 + a11 * d1;

    d3 = (1.0f - a23) * d2 + a23 * d3;
    d2 = (1.0f - a22) * d1 + a22 * d2;

    d3 = (1.0f - a33) * d2 + a33 * d3;

    out[idx] = d3;
  }
}

extern "C" void kernel_launch(void* const* d_in, const int* in_sizes, int n_in,
                              void* d_out, int out_size, void* d_ws, size_t ws_size,
                              hipStream_t stream) {
  const float* x = (const float*)d_in[0];
  const float* t = (const float*)d_in[1];
  const float* c = (const float*)d_in[2];
  float* out = (float*)d_out;

  (void)n_in; (void)out_size; (void)d_ws; (void)ws_size;

  const int B = in_sizes[0] / kN;  // query rows (65536 in the reference)

  int grid = 2048;                 // persistent blocks; 32 rows each at B=65536
  if (grid > B) grid = B;
  if (grid < 1) grid = 1;

  bspline_eval_gfx1250_kernel<<<grid, kN, 0, stream>>>(x, t, c, out, B);
}